// PillarFeatureNet_79989470921164
// MI455X (gfx1250) — compile-verified
//
#include <hip/hip_runtime.h>
#include <hip/hip_bf16.h>

typedef float v2f __attribute__((ext_vector_type(2)));
typedef float v8f __attribute__((ext_vector_type(8)));

#define IN_F 6
#define OUT_F 64
#define GRID_XY 512
#define BN_EPS 1e-5f

// ---------------------------------------------------------------------------
// Kernel 1: moments of x.  mom[0..5] = sum x_j ; mom[6..26] = upper-tri of
// sum x_j x_k (j<=k), index t(j,k) = j*6 - j*(j-1)/2 + (k-j).
// ---------------------------------------------------------------------------
__global__ __launch_bounds__(256) void pfn_moments_kernel(const float* __restrict__ x,
                                                          int n, float* __restrict__ mom) {
    float s[IN_F];
    float ss[21];
#pragma unroll
    for (int j = 0; j < IN_F; ++j) s[j] = 0.f;
#pragma unroll
    for (int t = 0; t < 21; ++t) ss[t] = 0.f;

    int tid = blockIdx.x * blockDim.x + threadIdx.x;
    int stride = gridDim.x * blockDim.x;
    for (int i = tid; i < n; i += stride) {
        const float* xp = x + (long long)i * IN_F;
        float v[IN_F];
#pragma unroll
        for (int j = 0; j < IN_F; ++j) v[j] = xp[j];
        int t = 0;
#pragma unroll
        for (int j = 0; j < IN_F; ++j) {
            s[j] += v[j];
#pragma unroll
            for (int k = j; k < IN_F; ++k) {
                ss[t] += v[j] * v[k];
                ++t;
            }
        }
    }

    // wave32 reduction
#pragma unroll
    for (int j = 0; j < IN_F; ++j)
        for (int off = 16; off > 0; off >>= 1) s[j] += __shfl_down(s[j], off, 32);
#pragma unroll
    for (int t = 0; t < 21; ++t)
        for (int off = 16; off > 0; off >>= 1) ss[t] += __shfl_down(ss[t], off, 32);

    if ((threadIdx.x & 31) == 0) {
#pragma unroll
        for (int j = 0; j < IN_F; ++j) atomicAdd(&mom[j], s[j]);
#pragma unroll
        for (int t = 0; t < 21; ++t) atomicAdd(&mom[IN_F + t], ss[t]);
    }
}

// ---------------------------------------------------------------------------
// Kernel 2: fold BatchNorm(batch stats) + bias into the linear layer.
// Bt[k*64 + c] = A_c * W[c][k]  (k in 0..7, rows 6..7 zero),
// bf[c] = A_c*(b_c - mean_c) + beta_c.
// ---------------------------------------------------------------------------
__global__ void pfn_fuse_kernel(const float* __restrict__ W, const float* __restrict__ b,
                                const float* __restrict__ gamma, const float* __restrict__ beta,
                                const float* __restrict__ mom, int n,
                                float* __restrict__ Bt, float* __restrict__ bf) {
    int c = threadIdx.x;
    if (c >= OUT_F) return;
    float s[IN_F];
#pragma unroll
    for (int j = 0; j < IN_F; ++j) s[j] = mom[j];

    float w[IN_F];
#pragma unroll
    for (int j = 0; j < IN_F; ++j) w[j] = W[c * IN_F + j];

    float wSw = 0.f;
    int t = 0;
#pragma unroll
    for (int j = 0; j < IN_F; ++j) {
#pragma unroll
        for (int k = j; k < IN_F; ++k) {
            float m = mom[IN_F + t];
            wSw += (j == k ? 1.f : 2.f) * w[j] * w[k] * m;
            ++t;
        }
    }
    float ws = 0.f;
#pragma unroll
    for (int j = 0; j < IN_F; ++j) ws += w[j] * s[j];

    float bc = b[c];
    float fn = (float)n;
    float sum_h = ws + fn * bc;
    float sumsq_h = wSw + 2.f * bc * ws + fn * bc * bc;
    float mean = sum_h / fn;
    float var = sumsq_h / fn - mean * mean;
    var = fmaxf(var, 0.f);
    float A = gamma[c] * rsqrtf(var + BN_EPS);

    bf[c] = A * (bc - mean) + beta[c];
#pragma unroll
    for (int k = 0; k < IN_F; ++k) Bt[k * OUT_F + c] = A * w[k];
    Bt[6 * OUT_F + c] = 0.f;  // K padding rows for the K=4..7 WMMA step
    Bt[7 * OUT_F + c] = 0.f;
}

// ---------------------------------------------------------------------------
// Kernel 3: fused linear(WMMA f32 16x16x4) + ReLU + scatter-add.
// One wave = one tile of 16 points x 64 channels (4 N-tiles, K = 4 + 4pad).
// A layout (16x4 f32): lanes0-15 -> K=0(v0),K=1(v1); lanes16-31 -> K=2,K=3.
// B layout (4x16 f32): lanes0-15 -> rows K=0(v0),K=1(v1); lanes16-31 -> K=2,K=3.
// C/D layout: vgpr r, lanes0-15 -> point r, chan=lane; lanes16-31 -> point r+8.
// Pillar base offsets are bpermute-broadcast ONCE per tile (8 ds ops), with
// invalid points encoded as a negative sentinel; full tiles take a
// branch-free path (the tail tile, if any, takes the guarded path).
// ---------------------------------------------------------------------------
__global__ __launch_bounds__(256) void pfn_main_kernel(const float* __restrict__ x,
                                                       const int* __restrict__ idx,
                                                       const float* __restrict__ Bt,
                                                       const float* __restrict__ bf,
                                                       float* __restrict__ grid, int n) {
    const int lane = threadIdx.x & 31;
    const int waveInBlk = threadIdx.x >> 5;
    const int half = lane >> 4;    // 0: lanes 0-15, 1: lanes 16-31
    const int nlo = lane & 15;

    const int wavesPerBlk = blockDim.x >> 5;
    const int gwave = blockIdx.x * wavesPerBlk + waveInBlk;
    const int nwaves = gridDim.x * wavesPerBlk;

    // Wave-invariant B operands (fused weights) and bias, loaded once.
    v2f B0[4], B1[4];
    float biasv[4];
#pragma unroll
    for (int nt = 0; nt < 4; ++nt) {
        const int col = nt * 16 + nlo;
        const int k0 = half * 2;       // K rows 0,1 (lanes<16) / 2,3 (lanes>=16)
        B0[nt].x = Bt[(k0 + 0) * OUT_F + col];
        B0[nt].y = Bt[(k0 + 1) * OUT_F + col];
        const int k1 = 4 + half * 2;   // K rows 4,5 / 6,7 (rows 6,7 are zero)
        B1[nt].x = Bt[(k1 + 0) * OUT_F + col];
        B1[nt].y = Bt[(k1 + 1) * OUT_F + col];
        biasv[nt] = bf[col];
    }

    const int tiles = (n + 15) >> 4;
    for (int t = gwave; t < tiles; t += nwaves) {
        const bool fullTile = (t * 16 + 16 <= n);   // wave-uniform
        int p = t * 16 + nlo;
        int pc = (p < n) ? p : (n - 1);

        // ---- A operand: 16 points x 6 features (K padded to 8) ----
        const float* xp = x + (long long)pc * IN_F;
        v2f a0, a1;
        const int ko = half * 2;       // K 0,1 or 2,3
        a0.x = xp[ko + 0];
        a0.y = xp[ko + 1];
        if (half == 0) {               // K 4,5
            a1.x = xp[4];
            a1.y = xp[5];
        } else {                       // K 6,7 (padding)
            a1.x = 0.f;
            a1.y = 0.f;
        }

        // ---- pillar base offsets: lanes 0-15 hold point nlo's offset;
        //      broadcast to the lanes holding each accumulator row (8 ds ops).
        int ip0 = idx[2 * pc + 0];
        int ip1 = idx[2 * pc + 1];
        int base = ((ip0 << 9) + ip1) << 6;        // (ix*512 + iy)*64, >= 0
        if (p >= n) base = (int)0xC0000000;        // negative sentinel (tail only)

        int baseR[8];
#pragma unroll
        for (int r = 0; r < 8; ++r)
            baseR[r] = __builtin_amdgcn_ds_bpermute((r + (half << 3)) << 2, base);

        if (fullTile) {
            // ---- hot path: branch-free wmma + relu + atomic scatter ----
#pragma unroll
            for (int nt = 0; nt < 4; ++nt) {
                v8f c;
#pragma unroll
                for (int i = 0; i < 8; ++i) c[i] = biasv[nt];
                c = __builtin_amdgcn_wmma_f32_16x16x4_f32(false, a0, false, B0[nt],
                                                          (short)0, c, false, false);
                c = __builtin_amdgcn_wmma_f32_16x16x4_f32(false, a1, false, B1[nt],
                                                          (short)0, c, false, false);
                const int ch = nt * 16 + nlo;
#pragma unroll
                for (int r = 0; r < 8; ++r)
                    atomicAdd(&grid[baseR[r] + ch], fmaxf(c[r], 0.f));
            }
        } else {
            // ---- tail tile: guard each row by the sentinel ----
#pragma unroll
            for (int nt = 0; nt < 4; ++nt) {
                v8f c;
#pragma unroll
                for (int i = 0; i < 8; ++i) c[i] = biasv[nt];
                c = __builtin_amdgcn_wmma_f32_16x16x4_f32(false, a0, false, B0[nt],
                                                          (short)0, c, false, false);
                c = __builtin_amdgcn_wmma_f32_16x16x4_f32(false, a1, false, B1[nt],
                                                          (short)0, c, false, false);
                const int ch = nt * 16 + nlo;
#pragma unroll
                for (int r = 0; r < 8; ++r)
                    if (baseR[r] >= 0) atomicAdd(&grid[baseR[r] + ch], fmaxf(c[r], 0.f));
            }
        }
    }
}

// ---------------------------------------------------------------------------
extern "C" void kernel_launch(void* const* d_in, const int* in_sizes, int n_in,
                              void* d_out, int out_size, void* d_ws, size_t ws_size,
                              hipStream_t stream) {
    const float* x     = (const float*)d_in[0];
    const int*   idx   = (const int*)d_in[1];
    const float* W     = (const float*)d_in[2];
    const float* b     = (const float*)d_in[3];
    const float* gamma = (const float*)d_in[4];
    const float* beta  = (const float*)d_in[5];
    float* grid = (float*)d_out;

    const int n = in_sizes[0] / IN_F;
    if (n <= 0) return;

    float* mom = (float*)d_ws;      // 27 floats used (64 reserved)
    float* Bt  = mom + 64;          // 8 x 64 fused weights (K-major, K padded)
    float* bf  = Bt + 8 * OUT_F;    // 64 fused biases

    // zero output grid (atomic accumulation target) and moments
    hipMemsetAsync(d_out, 0, (size_t)out_size * sizeof(float), stream);
    hipMemsetAsync(mom, 0, 64 * sizeof(float), stream);

    pfn_moments_kernel<<<512, 256, 0, stream>>>(x, n, mom);
    pfn_fuse_kernel<<<1, 64, 0, stream>>>(W, b, gamma, beta, mom, n, Bt, bf);

    const int tiles = (n + 15) >> 4;
    int blocks = (tiles + 7) / 8;            // 8 waves per block, 1 tile per wave
    if (blocks > 16384) blocks = 16384;      // grid-stride covers the rest
    pfn_main_kernel<<<blocks, 256, 0, stream>>>(x, idx, Bt, bf, grid, n);
}